// TopDown_56702158241997
// MI455X (gfx1250) — compile-verified
//
#include <hip/hip_runtime.h>
#include <math.h>

#if defined(__AMDGCN__) && __has_builtin(__builtin_amdgcn_global_load_async_to_lds_b128) && \
    __has_builtin(__builtin_amdgcn_global_load_async_to_lds_b32) && \
    __has_builtin(__builtin_amdgcn_s_wait_asynccnt)
#define HAVE_ASYNC 1
#else
#define HAVE_ASYNC 0
#endif

typedef __attribute__((ext_vector_type(2))) float v2f;
typedef __attribute__((ext_vector_type(4))) float v4f;
typedef __attribute__((ext_vector_type(8))) float v8f;
typedef int vec4i __attribute__((vector_size(16)));  // matches builtin proto

#define AS_GLOBAL __attribute__((address_space(1)))
#define AS_LDS    __attribute__((address_space(3)))

__device__ __forceinline__ v8f wmma_f32_k4(v2f a, v2f b, v8f c) {
  // D(16x16,f32) = A(16x4) * B(4x16) + C   (wave32, f32 WMMA)
  return __builtin_amdgcn_wmma_f32_16x16x4_f32(false, a, false, b, (short)0, c,
                                               false, false);
}

// ---- global -> LDS staging helpers (async-to-LDS when available) ----------
__device__ __forceinline__ void cp_b128(const float* g, float* l) {
#if HAVE_ASYNC
  __builtin_amdgcn_global_load_async_to_lds_b128(
      (AS_GLOBAL vec4i*)g, (AS_LDS vec4i*)l, 0, 0);
#else
  *(v4f*)l = *(const v4f*)g;
#endif
}
__device__ __forceinline__ void cp_b32(const float* g, float* l) {
#if HAVE_ASYNC
  __builtin_amdgcn_global_load_async_to_lds_b32(
      (AS_GLOBAL int*)g, (AS_LDS int*)l, 0, 0);
#else
  *l = *g;
#endif
}
__device__ __forceinline__ void staging_fence() {
#if HAVE_ASYNC
  __builtin_amdgcn_s_wait_asynccnt(0);
#endif
  __syncthreads();
}

#define KC 64   // K-chunk
#define SA 68   // LDS row stride (floats): bank-conflict-free, 16B-aligned
#define SB 68

// ---------------------------------------------------------------------------
// s = x + nearest-neighbor-up2(t)
// ---------------------------------------------------------------------------
__global__ void k_upadd(const float* __restrict__ x, const float* __restrict__ t,
                        float* __restrict__ s, int H, int W, int total) {
  int i = blockIdx.x * blockDim.x + threadIdx.x;
  if (i >= total) return;
  int w = i % W;
  int h = (i / W) % H;
  int bc = i / (W * H);
  int Hh = H >> 1, Wh = W >> 1;
  s[i] = x[i] + t[((size_t)bc * Hh + (h >> 1)) * Wh + (w >> 1)];
}

// ---------------------------------------------------------------------------
// Implicit-GEMM 3x3 conv (pad=1), LDS double-buffered pipeline.
// Workgroup: 256 thr = 8 waves; tile M=64 x N=32; wave = 16x16 subtile.
// A staged via async-to-LDS; B (im2col with value select) pipelined via regs.
// ---------------------------------------------------------------------------
__global__ __launch_bounds__(256) void k_conv3x3_wmma(
    const float* __restrict__ in, const float* __restrict__ wt,
    const float* __restrict__ bias, float* __restrict__ out,
    int Cout, int Cin, int H, int W) {
  __shared__ float As[2][64 * SA];
  __shared__ float Bs[2][32 * SB];
  const int N = H * W;
  const int K = Cin * 9;
  const int tid = threadIdx.x;
  const int lane = tid & 31;
  const int wave = tid >> 5;
  const int waveM = wave >> 1;  // 0..3
  const int waveN = wave & 1;   // 0..1
  const bool hi = lane >= 16;
  const int l = lane & 15;
  const int koff = hi ? 2 : 0;
  const int nb = blockIdx.x * 32;
  const int mb = blockIdx.y * 64;
  const int bb = blockIdx.z;

  const float* __restrict__ inb = in + (size_t)bb * Cin * N;

  // ---- chunk-invariant staging coordinates ----
  const int a_kq = (tid & 15) << 2;  // k-quad within chunk
  int a_row[4], a_mrow[4];
#pragma unroll
  for (int j = 0; j < 4; ++j) {
    a_row[j] = (tid >> 4) + 16 * j;
    int mr = mb + a_row[j];
    a_mrow[j] = mr > Cout - 1 ? Cout - 1 : mr;  // pad rows (stores guarded)
  }
  const int b_nn = tid & 31;
  int b_n = nb + b_nn;
  if (b_n > N - 1) b_n = N - 1;
  const int b_h = b_n / W;
  const int b_w = b_n - b_h * W;
  int b_kk[8], b_ldso[8];
#pragma unroll
  for (int j = 0; j < 8; ++j) {
    b_kk[j] = (tid >> 5) + 8 * j;
    b_ldso[j] = b_nn * SB + b_kk[j];
  }

  v8f acc;
#pragma unroll
  for (int r = 0; r < 8; ++r) {
    int m = mb + waveM * 16 + r + (hi ? 8 : 0);
    acc[r] = (m < Cout) ? bias[m] : 0.0f;
  }

  float bv[8];
  bool bok[8];

  auto stageA = [&](int kc, int buf) {
#pragma unroll
    for (int j = 0; j < 4; ++j)
      cp_b128(wt + (size_t)a_mrow[j] * K + kc + a_kq, &As[buf][a_row[j] * SA + a_kq]);
  };
  auto loadB = [&](int kc) {  // branchless im2col gather into regs
#pragma unroll
    for (int j = 0; j < 8; ++j) {
      int k = kc + b_kk[j];
      int c = k / 9;
      int r9 = k - c * 9;
      int ky = r9 / 3;
      int kx = r9 - ky * 3;
      int y = b_h + ky - 1;
      int x = b_w + kx - 1;
      bok[j] = ((unsigned)y < (unsigned)H) && ((unsigned)x < (unsigned)W);
      int yc = y < 0 ? 0 : (y > H - 1 ? H - 1 : y);
      int xc = x < 0 ? 0 : (x > W - 1 ? W - 1 : x);
      bv[j] = inb[(size_t)c * N + yc * W + xc];  // clamped, always-safe load
    }
  };
  auto storeB = [&](int buf) {
#pragma unroll
    for (int j = 0; j < 8; ++j)
      Bs[buf][b_ldso[j]] = bok[j] ? bv[j] : 0.0f;  // v_cndmask + ds_store
  };
  auto compute = [&](int buf) {
    const float* arow = &As[buf][(waveM * 16 + l) * SA + koff];
    const float* brow = &Bs[buf][(waveN * 16 + l) * SB + koff];
#pragma unroll
    for (int k2 = 0; k2 < KC; k2 += 4) {
      v2f a = *(const v2f*)(arow + k2);
      v2f b = *(const v2f*)(brow + k2);
      acc = wmma_f32_k4(a, b, acc);
    }
  };

  const int nch = K / KC;  // 36
  // prologue: stage chunk 0 into buffer 0
  stageA(0, 0);
  loadB(0);
  storeB(0);
  staging_fence();
  int buf = 0;
  for (int ch = 0; ch < nch; ++ch) {
    if (ch + 1 < nch) {          // overlap: issue next chunk's loads first
      stageA((ch + 1) * KC, buf ^ 1);  // async-to-LDS, no wait yet
      loadB((ch + 1) * KC);            // global loads in flight during compute
    }
    compute(buf);                // WMMA from current buffer
    if (ch + 1 < nch) storeB(buf ^ 1);
    staging_fence();             // s_wait_asynccnt 0 + barrier
    buf ^= 1;
  }

  float* __restrict__ ob = out + (size_t)bb * Cout * N;
  const int n = nb + waveN * 16 + l;
#pragma unroll
  for (int r = 0; r < 8; ++r) {
    int m = mb + waveM * 16 + r + (hi ? 8 : 0);
    if (m < Cout && n < N) ob[(size_t)m * N + n] = acc[r];
  }
}

// ---------------------------------------------------------------------------
// DCNv2 column builder: cols[b][c*9+k][n] = mask * bilinear(s[b,c], p(k,n)+off)
// ---------------------------------------------------------------------------
__global__ void k_build_cols(const float* __restrict__ s,
                             const float* __restrict__ off,
                             float* __restrict__ cols,
                             int Cin, int H, int W, int total) {
  const int N = H * W;
  int tid = blockIdx.x * blockDim.x + threadIdx.x;
  if (tid >= total) return;
  int n = tid % N;
  int k = (tid / N) % 9;
  int b = tid / (9 * N);

  const float* __restrict__ offb = off + (size_t)b * 27 * N;
  float dy = offb[(size_t)(2 * k) * N + n];
  float dx = offb[(size_t)(2 * k + 1) * N + n];
  float mk = offb[(size_t)(18 + k) * N + n];
  mk = 1.0f / (1.0f + expf(-mk));

  int h = n / W;
  int w = n - h * W;
  float py = (float)h - 1.0f + (float)(k / 3) + dy;
  float px = (float)w - 1.0f + (float)(k % 3) + dx;
  float y0f = floorf(py), x0f = floorf(px);
  float wy = py - y0f, wx = px - x0f;
  int y0 = (int)y0f, x0 = (int)x0f;
  int y1 = y0 + 1, x1 = x0 + 1;

  int y0c = y0 < 0 ? 0 : (y0 > H - 1 ? H - 1 : y0);
  int y1c = y1 < 0 ? 0 : (y1 > H - 1 ? H - 1 : y1);
  int x0c = x0 < 0 ? 0 : (x0 > W - 1 ? W - 1 : x0);
  int x1c = x1 < 0 ? 0 : (x1 > W - 1 ? W - 1 : x1);
  bool vy0 = (y0 >= 0) && (y0 < H);
  bool vy1 = (y1 >= 0) && (y1 < H);
  bool vx0 = (x0 >= 0) && (x0 < W);
  bool vx1 = (x1 >= 0) && (x1 < W);
  int i00 = y0c * W + x0c;
  int i01 = y0c * W + x1c;
  int i10 = y1c * W + x0c;
  int i11 = y1c * W + x1c;
  float w00 = (vy0 && vx0) ? (1.0f - wy) * (1.0f - wx) * mk : 0.0f;
  float w01 = (vy0 && vx1) ? (1.0f - wy) * wx * mk : 0.0f;
  float w10 = (vy1 && vx0) ? wy * (1.0f - wx) * mk : 0.0f;
  float w11 = (vy1 && vx1) ? wy * wx * mk : 0.0f;

  const float* __restrict__ sb = s + (size_t)b * Cin * N;
  float* __restrict__ cp = cols + ((size_t)b * Cin * 9 + k) * N + n;
  for (int c = 0; c < Cin; ++c) {
    const float* __restrict__ p = sb + (size_t)c * N;
    float v = w00 * p[i00] + w01 * p[i01] + w10 * p[i10] + w11 * p[i11];
    cp[(size_t)c * 9 * N] = v;
  }
}

// ---------------------------------------------------------------------------
// Dense GEMM over precomputed cols, double-buffered, fully async staging.
// ---------------------------------------------------------------------------
__global__ __launch_bounds__(256) void k_gemm_cols_wmma(
    const float* __restrict__ cols, const float* __restrict__ wt,
    const float* __restrict__ bias, float* __restrict__ out,
    int Cout, int K, int N) {
  __shared__ float As[2][64 * SA];
  __shared__ float Bs[2][32 * SB];
  const int tid = threadIdx.x;
  const int lane = tid & 31;
  const int wave = tid >> 5;
  const int waveM = wave >> 1;
  const int waveN = wave & 1;
  const bool hi = lane >= 16;
  const int l = lane & 15;
  const int koff = hi ? 2 : 0;
  const int nb = blockIdx.x * 32;
  const int mb = blockIdx.y * 64;
  const int bb = blockIdx.z;

  const float* __restrict__ cb = cols + (size_t)bb * K * N;

  const int a_kq = (tid & 15) << 2;
  int a_row[4], a_mrow[4];
#pragma unroll
  for (int j = 0; j < 4; ++j) {
    a_row[j] = (tid >> 4) + 16 * j;
    int mr = mb + a_row[j];
    a_mrow[j] = mr > Cout - 1 ? Cout - 1 : mr;
  }
  const int b_nn = tid & 31;
  int b_n = nb + b_nn;
  if (b_n > N - 1) b_n = N - 1;
  int b_kk[8], b_ldso[8];
#pragma unroll
  for (int j = 0; j < 8; ++j) {
    b_kk[j] = (tid >> 5) + 8 * j;
    b_ldso[j] = b_nn * SB + b_kk[j];
  }

  v8f acc;
#pragma unroll
  for (int r = 0; r < 8; ++r) {
    int m = mb + waveM * 16 + r + (hi ? 8 : 0);
    acc[r] = (m < Cout) ? bias[m] : 0.0f;
  }

  auto stageA = [&](int kc, int buf) {
#pragma unroll
    for (int j = 0; j < 4; ++j)
      cp_b128(wt + (size_t)a_mrow[j] * K + kc + a_kq, &As[buf][a_row[j] * SA + a_kq]);
  };
  auto stageB = [&](int kc, int buf) {  // transposed scatter, async per element
#pragma unroll
    for (int j = 0; j < 8; ++j)
      cp_b32(cb + (size_t)(kc + b_kk[j]) * N + b_n, &Bs[buf][b_ldso[j]]);
  };
  auto compute = [&](int buf) {
    const float* arow = &As[buf][(waveM * 16 + l) * SA + koff];
    const float* brow = &Bs[buf][(waveN * 16 + l) * SB + koff];
#pragma unroll
    for (int k2 = 0; k2 < KC; k2 += 4) {
      v2f a = *(const v2f*)(arow + k2);
      v2f b = *(const v2f*)(brow + k2);
      acc = wmma_f32_k4(a, b, acc);
    }
  };

  const int nch = K / KC;  // 36
  stageA(0, 0);
  stageB(0, 0);
  staging_fence();
  int buf = 0;
  for (int ch = 0; ch < nch; ++ch) {
    if (ch + 1 < nch) {  // overlap next chunk's async staging with compute
      stageA((ch + 1) * KC, buf ^ 1);
      stageB((ch + 1) * KC, buf ^ 1);
    }
    compute(buf);
    staging_fence();
    buf ^= 1;
  }

  float* __restrict__ ob = out + (size_t)bb * Cout * N;
  const int n = nb + waveN * 16 + l;
#pragma unroll
  for (int r = 0; r < 8; ++r) {
    int m = mb + waveM * 16 + r + (hi ? 8 : 0);
    if (m < Cout && n < N) ob[(size_t)m * N + n] = acc[r];
  }
}

// ---------------------------------------------------------------------------
extern "C" void kernel_launch(void* const* d_in, const int* in_sizes, int n_in,
                              void* d_out, int out_size, void* d_ws, size_t ws_size,
                              hipStream_t stream) {
  // setup_inputs() dict order:
  const float* x0 = (const float*)d_in[0];
  const float* x1 = (const float*)d_in[1];
  const float* x2 = (const float*)d_in[2];
  const float* x3 = (const float*)d_in[3];
  const float* w0 = (const float*)d_in[4];
  const float* b0 = (const float*)d_in[5];
  const float* w1 = (const float*)d_in[6];
  const float* b1 = (const float*)d_in[7];
  const float* w2 = (const float*)d_in[8];
  const float* b2 = (const float*)d_in[9];
  const float* w3 = (const float*)d_in[10];
  const float* b3 = (const float*)d_in[11];
  const float* ow1 = (const float*)d_in[12];
  const float* ob1 = (const float*)d_in[13];
  const float* ow2 = (const float*)d_in[14];
  const float* ob2 = (const float*)d_in[15];
  const float* ow3 = (const float*)d_in[16];
  const float* ob3 = (const float*)d_in[17];

  float* out = (float*)d_out;
  float* ws = (float*)d_ws;

  const int Bn = 2, C = 256, K = C * 9;

  float* t0 = out;
  float* t1 = t0 + (size_t)Bn * C * 96 * 96;
  float* t2 = t1 + (size_t)Bn * C * 48 * 48;
  float* t3 = t2 + (size_t)Bn * C * 24 * 24;

  float* s = ws;                                  // up to 2*256*96*96
  float* off = s + (size_t)Bn * C * 96 * 96;      // up to 2*27*48*48
  float* cols = off + (size_t)Bn * 27 * 48 * 48;  // up to 2*2304*2304

  // ---- Level 3 (12x12): t3 = mdconv(x3) ----
  {
    const int H = 12, W = 12, N = H * W;
    k_conv3x3_wmma<<<dim3((N + 31) / 32, 1, Bn), 256, 0, stream>>>(x3, ow3, ob3, off, 27, C, H, W);
    int tot = Bn * 9 * N;
    k_build_cols<<<(tot + 255) / 256, 256, 0, stream>>>(x3, off, cols, C, H, W, tot);
    k_gemm_cols_wmma<<<dim3((N + 31) / 32, C / 64, Bn), 256, 0, stream>>>(cols, w3, b3, t3, C, K, N);
  }
  // ---- Level 2 (24x24): t2 = mdconv(x2 + up2(t3)) ----
  {
    const int H = 24, W = 24, N = H * W;
    int tot = Bn * C * N;
    k_upadd<<<(tot + 255) / 256, 256, 0, stream>>>(x2, t3, s, H, W, tot);
    k_conv3x3_wmma<<<dim3((N + 31) / 32, 1, Bn), 256, 0, stream>>>(s, ow2, ob2, off, 27, C, H, W);
    int tc = Bn * 9 * N;
    k_build_cols<<<(tc + 255) / 256, 256, 0, stream>>>(s, off, cols, C, H, W, tc);
    k_gemm_cols_wmma<<<dim3((N + 31) / 32, C / 64, Bn), 256, 0, stream>>>(cols, w2, b2, t2, C, K, N);
  }
  // ---- Level 1 (48x48): t1 = mdconv(x1 + up2(t2)) ----
  {
    const int H = 48, W = 48, N = H * W;
    int tot = Bn * C * N;
    k_upadd<<<(tot + 255) / 256, 256, 0, stream>>>(x1, t2, s, H, W, tot);
    k_conv3x3_wmma<<<dim3((N + 31) / 32, 1, Bn), 256, 0, stream>>>(s, ow1, ob1, off, 27, C, H, W);
    int tc = Bn * 9 * N;
    k_build_cols<<<(tc + 255) / 256, 256, 0, stream>>>(s, off, cols, C, H, W, tc);
    k_gemm_cols_wmma<<<dim3((N + 31) / 32, C / 64, Bn), 256, 0, stream>>>(cols, w1, b1, t1, C, K, N);
  }
  // ---- Level 0 (96x96): t0 = conv(x0 + up2(t1)) ----
  {
    const int H = 96, W = 96, N = H * W;
    int tot = Bn * C * N;
    k_upadd<<<(tot + 255) / 256, 256, 0, stream>>>(x0, t1, s, H, W, tot);
    k_conv3x3_wmma<<<dim3((N + 31) / 32, C / 64, Bn), 256, 0, stream>>>(s, w0, b0, t0, C, C, H, W);
  }
}